// GNN_30296699306729
// MI455X (gfx1250) — compile-verified
//
#include <hip/hip_runtime.h>
#include <math.h>
#include <stdint.h>

typedef __attribute__((ext_vector_type(2))) float v2f;
typedef __attribute__((ext_vector_type(8))) float v8f;

#define B_  16
#define P_  12
#define N_  20000
#define E_  320000

// folded-constant region layout (float offsets)
#define C_AZ 0      // Az[2][12]
#define C_CZ 24     // cz[12]
#define C_AH 36     // Ah[2][12]
#define C_CH 60     // ch[12]
#define C_PR 72     // probs[12]
#define C_TOT 84

// ---------------------------------------------------------------- GAIN fill
__global__ void fill_kernel(const float* __restrict__ x, const float* __restrict__ m,
                            const float* __restrict__ nz, float* __restrict__ xf, int total) {
    int i = blockIdx.x * blockDim.x + threadIdx.x;
    if (i < total) {
        float mm = m[i];
        xf[i] = mm * x[i] + (1.0f - mm) * nz[i];
    }
}

// ------------------------------------------------------------- degree/dinv
__global__ void deg_kernel(const int* __restrict__ ei, const float* __restrict__ w,
                           float* __restrict__ deg, int E) {
    int e = blockIdx.x * blockDim.x + threadIdx.x;
    if (e < E) atomicAdd(&deg[ei[E + e]], w[e]);   // dst = edge_index[1][e]
}

__global__ void dinv_kernel(float* __restrict__ deg, int N) {
    int n = blockIdx.x * blockDim.x + threadIdx.x;
    if (n < N) {
        float d = deg[n] + 1.0f;                   // self-loop weight 1 at dst
        deg[n] = (d > 0.0f) ? rsqrtf(d) : 0.0f;    // overwrite in place with dinv
    }
}

__global__ void norm_kernel(const int* __restrict__ ei, const float* __restrict__ w,
                            const float* __restrict__ dinv, float* __restrict__ nrm, int E) {
    int e = blockIdx.x * blockDim.x + threadIdx.x;
    if (e < E) nrm[e] = dinv[ei[e]] * w[e] * dinv[ei[E + e]];
}

// ----------------- edge scatter: one thread per edge, loop all (b,p) slabs.
// Edge index / norm read exactly once; atomics (the bottleneck) unchanged.
__global__ void scatter_kernel(const int* __restrict__ ei, const float* __restrict__ nrm,
                               const float* __restrict__ xf, const float* __restrict__ mask,
                               float* __restrict__ agg, int E) {
    int e = blockIdx.x * blockDim.x + threadIdx.x;
    if (e >= E) return;
    int s = ei[e], d = ei[E + e];
    float nw = nrm[e];
    for (int bp = 0; bp < B_ * P_; ++bp) {
        size_t base = (size_t)bp * N_;
        if (bp + 1 < B_ * P_) {                     // prefetch next slab's gather lines
            __builtin_prefetch(&xf[base + N_ + s], 0, 1);
            __builtin_prefetch(&mask[base + N_ + s], 0, 1);
        }
        float vx = nw * xf[base + s];
        float vm = nw * mask[base + s];
        float* out = agg + (base + (size_t)d) * 2;
        atomicAdd(out,     vx);
        atomicAdd(out + 1, vm);
    }
}

// -------------------------------------------- fold GRU weights (H0 == 0 path)
__global__ void consts_kernel(const float* __restrict__ Wz, const float* __restrict__ bz,
                              const float* __restrict__ Wh, const float* __restrict__ bh,
                              const float* __restrict__ Lzw, const float* __restrict__ Lzb,
                              const float* __restrict__ Lhw, const float* __restrict__ Lhb,
                              const float* __restrict__ att, float* __restrict__ C) {
    int j = threadIdx.x;
    if (j < P_) {
        float az0 = 0.f, az1 = 0.f, ah0 = 0.f, ah1 = 0.f;
        float cz = Lzb[j], ch = Lhb[j];
        for (int k = 0; k < P_; ++k) {
            float lz = Lzw[k * P_ + j], lh = Lhw[k * P_ + j];
            az0 += Wz[k] * lz;  az1 += Wz[P_ + k] * lz;
            ah0 += Wh[k] * lh;  ah1 += Wh[P_ + k] * lh;
            cz  += bz[k] * lz;  ch  += bh[k] * lh;
        }
        C[C_AZ + j] = az0;  C[C_AZ + P_ + j] = az1;  C[C_CZ + j] = cz;
        C[C_AH + j] = ah0;  C[C_AH + P_ + j] = ah1;  C[C_CH + j] = ch;
        float mx = att[0];
        for (int k = 1; k < P_; ++k) mx = fmaxf(mx, att[k]);
        float sum = 0.f;
        for (int k = 0; k < P_; ++k) sum += __expf(att[k] - mx);
        C[C_PR + j] = __expf(att[j] - mx) / sum;
    }
}

// ------------------- per-node GRU collapse, loop periods, accumulate H_accum
__global__ void node_kernel(const float* __restrict__ agg, const float* __restrict__ xf,
                            const float* __restrict__ mask, const float* __restrict__ dinv,
                            const float* __restrict__ C, float* __restrict__ Hacc) {
    __shared__ float cs[C_TOT];
    if (threadIdx.x < C_TOT) cs[threadIdx.x] = C[threadIdx.x];
    __syncthreads();
    int i = blockIdx.x * blockDim.x + threadIdx.x;     // b*N + n
    if (i >= B_ * N_) return;
    int b = i / N_, n = i - b * N_;
    float selfw = dinv[n]; selfw *= selfw;             // dinv[n] * 1 * dinv[n]
    float acc[P_];
#pragma unroll
    for (int j = 0; j < P_; ++j) acc[j] = 0.f;
    for (int p = 0; p < P_; ++p) {
        size_t bp = (size_t)(b * P_ + p);
        size_t gi = bp * N_ + (size_t)n;
        float ax = agg[gi * 2]     + selfw * xf[gi];
        float am = agg[gi * 2 + 1] + selfw * mask[gi];
        float pr = cs[C_PR + p];
#pragma unroll
        for (int j = 0; j < P_; ++j) {
            float z  = 1.f / (1.f + __expf(-(ax * cs[C_AZ + j] + am * cs[C_AZ + P_ + j] + cs[C_CZ + j])));
            float ht = tanhf(ax * cs[C_AH + j] + am * cs[C_AH + P_ + j] + cs[C_CH + j]);
            acc[j] += pr * (1.f - z) * ht;             // H = (1-Z)*Ht  (H0 == 0)
        }
    }
    float* out = Hacc + (size_t)i * P_;
#pragma unroll
    for (int j = 0; j < P_; ++j) out[j] = acc[j];
}

// ---------------- MLP head via fp32 WMMA: relu -> @W1+b1 -> relu -> @W2+b2 -> sigmoid
// One wave owns a 16-node x 16-feature tile; K=12 = 3 chained 16x16x4 WMMAs per GEMM.
// Weights/biases are staged zero-padded into LDS (real elements via
// global_load_async_to_lds_b32, ASYNCcnt-fenced) so the WMMA stream has no
// exec-mask manipulation and no per-wave global weight traffic.
__global__ void __launch_bounds__(128)
mlp_wmma_kernel(const float* __restrict__ Hacc,
                const float* __restrict__ W1, const float* __restrict__ b1,
                const float* __restrict__ W2, const float* __restrict__ b2,
                const float* __restrict__ xf, const float* __restrict__ mask,
                float* __restrict__ res, float* __restrict__ imp) {
    __shared__ float lds[4][16 * 17];                  // per-wave transpose tile (pad 17)
    __shared__ float sW1[P_ * 16];                     // [12][16], cols 12..15 = 0
    __shared__ float sW2[P_ * 16];
    __shared__ float sb[32];                           // b1 padded | b2 padded

    int t = threadIdx.x;
    if (t < 32) {
        int col = t & 15;
        const float* bsrc = (t < 16) ? b1 : b2;
        sb[t] = (col < P_) ? bsrc[col] : 0.f;
    }
    for (int idx = t; idx < P_ * 16; idx += 128) {
        int row = idx >> 4, col = idx & 15;
        if (col < P_) {
            uint64_t ga1 = (uint64_t)(uintptr_t)(W1 + row * P_ + col);
            uint32_t la1 = (uint32_t)(uintptr_t)(&sW1[idx]);
            asm volatile("global_load_async_to_lds_b32 %0, %1, off"
                         :: "v"(la1), "v"(ga1) : "memory");
            uint64_t ga2 = (uint64_t)(uintptr_t)(W2 + row * P_ + col);
            uint32_t la2 = (uint32_t)(uintptr_t)(&sW2[idx]);
            asm volatile("global_load_async_to_lds_b32 %0, %1, off"
                         :: "v"(la2), "v"(ga2) : "memory");
        } else {
            sW1[idx] = 0.f;
            sW2[idx] = 0.f;
        }
    }
    asm volatile("s_wait_asynccnt 0x0" ::: "memory");  // my async transfers done
    __syncthreads();                                    // visible block-wide

    int wave = t >> 5;
    int lane = t & 31;
    int tile = blockIdx.x * 4 + wave;                  // 0 .. B*(N/16)-1
    int b = tile / (N_ / 16);
    int nodeBase = (tile - b * (N_ / 16)) * 16;
    int g  = lane >> 4;                                // half-wave: A holds K=2g..2g+1 of slice
    int mn = lane & 15;                                // A: row m ; B/C/D: col n

    // ---- GEMM1: D = relu(Hacc) @ W1 + b1   (cols >= 12 are exactly 0)
    v8f c;
    float bias1 = sb[mn];
#pragma unroll
    for (int r = 0; r < 8; ++r) c[r] = bias1;
    const float* hrow = Hacc + ((size_t)b * N_ + nodeBase + mn) * P_;
#pragma unroll
    for (int s = 0; s < 3; ++s) {
        int k0 = 4 * s + 2 * g;                        // k0, k0+1 in [0,11]
        v2f a, w;
        a.x = fmaxf(hrow[k0], 0.f);
        a.y = fmaxf(hrow[k0 + 1], 0.f);
        w.x = sW1[k0 * 16 + mn];
        w.y = sW1[(k0 + 1) * 16 + mn];
        c = __builtin_amdgcn_wmma_f32_16x16x4_f32(false, a, false, w, (short)0, c, false, false);
    }
    // relu + transpose through LDS: D lane(g,n) reg r -> element [row r+8g][col n]
    float* T = lds[wave];
#pragma unroll
    for (int r = 0; r < 8; ++r) T[(r + 8 * g) * 17 + mn] = fmaxf(c[r], 0.f);
    __syncthreads();

    // ---- GEMM2: D2 = T @ W2 + b2
    v8f c2;
    float bias2 = sb[16 + mn];
#pragma unroll
    for (int r = 0; r < 8; ++r) c2[r] = bias2;
#pragma unroll
    for (int s = 0; s < 3; ++s) {
        int k0 = 4 * s + 2 * g;
        v2f a, w;
        a.x = T[mn * 17 + k0];
        a.y = T[mn * 17 + k0 + 1];
        w.x = sW2[k0 * 16 + mn];
        w.y = sW2[(k0 + 1) * 16 + mn];
        c2 = __builtin_amdgcn_wmma_f32_16x16x4_f32(false, a, false, w, (short)0, c2, false, false);
    }

    // ---- sigmoid + write res / imputation (both [B,P,N]); lane col mn is period p
    if (mn < P_) {
        int p = mn;
#pragma unroll
        for (int r = 0; r < 8; ++r) {
            int node = nodeBase + r + 8 * g;
            float v = 1.f / (1.f + __expf(-c2[r]));
            size_t idx = ((size_t)b * P_ + p) * N_ + (size_t)node;
            float m = mask[idx];
            imp[idx] = v;
            res[idx] = m * xf[idx] + (1.f - m) * v;
        }
    }
}

// ---------------------------------------------------------------------------
extern "C" void kernel_launch(void* const* d_in, const int* in_sizes, int n_in,
                              void* d_out, int out_size, void* d_ws, size_t ws_size,
                              hipStream_t stream) {
    const float* x     = (const float*)d_in[0];
    const float* mask  = (const float*)d_in[1];
    const float* noise = (const float*)d_in[2];
    const int*   ei    = (const int*)d_in[3];
    const float* ew    = (const float*)d_in[4];
    const float* Wz    = (const float*)d_in[5];
    const float* bz    = (const float*)d_in[6];
    // d_in[7]=Wr, d_in[8]=br  : dead (H0 == 0)
    const float* Wh    = (const float*)d_in[9];
    const float* bh    = (const float*)d_in[10];
    const float* Lzw   = (const float*)d_in[11];
    const float* Lzb   = (const float*)d_in[12];
    // d_in[13]=Lr_w, d_in[14]=Lr_b : dead
    const float* Lhw   = (const float*)d_in[15];
    const float* Lhb   = (const float*)d_in[16];
    const float* att   = (const float*)d_in[17];
    const float* W1    = (const float*)d_in[18];
    const float* b1    = (const float*)d_in[19];
    const float* W2    = (const float*)d_in[20];
    const float* b2    = (const float*)d_in[21];
    (void)in_sizes; (void)n_in; (void)out_size; (void)ws_size;

    float* res = (float*)d_out;
    float* imp = res + (size_t)B_ * P_ * N_;

    char* ws = (char*)d_ws;
    size_t o = 0;
    float* xf   = (float*)(ws + o); o += (size_t)B_ * P_ * N_ * 4;      // 15.36 MB
    float* deg  = (float*)(ws + o); o += (size_t)N_ * 4;                // dinv after pass 2
    float* nrm  = (float*)(ws + o); o += (size_t)E_ * 4;
    float* agg  = (float*)(ws + o); o += (size_t)B_ * P_ * N_ * 2 * 4;  // 30.72 MB
    float* Hacc = (float*)(ws + o); o += (size_t)B_ * N_ * P_ * 4;      // 15.36 MB
    float* C    = (float*)(ws + o); o += (size_t)C_TOT * 4;

    hipMemsetAsync(deg, 0, (size_t)N_ * 4, stream);
    hipMemsetAsync(agg, 0, (size_t)B_ * P_ * N_ * 2 * 4, stream);

    int total = B_ * P_ * N_;
    fill_kernel<<<(total + 255) / 256, 256, 0, stream>>>(x, mask, noise, xf, total);
    deg_kernel<<<(E_ + 255) / 256, 256, 0, stream>>>(ei, ew, deg, E_);
    dinv_kernel<<<(N_ + 255) / 256, 256, 0, stream>>>(deg, N_);
    norm_kernel<<<(E_ + 255) / 256, 256, 0, stream>>>(ei, ew, deg, nrm, E_);
    scatter_kernel<<<(E_ + 255) / 256, 256, 0, stream>>>(ei, nrm, xf, mask, agg, E_);
    consts_kernel<<<1, 32, 0, stream>>>(Wz, bz, Wh, bh, Lzw, Lzb, Lhw, Lhb, att, C);
    node_kernel<<<(B_ * N_ + 255) / 256, 256, 0, stream>>>(agg, xf, mask, deg, C, Hacc);
    mlp_wmma_kernel<<<(B_ * (N_ / 16)) / 4, 128, 0, stream>>>(Hacc, W1, b1, W2, b2, xf, mask, res, imp);
}